// Decoding_77841987272832
// MI455X (gfx1250) — compile-verified
//
#include <hip/hip_runtime.h>
#include <hip/hip_bf16.h>
#include <math.h>

// ---------------------------------------------------------------------------
// Shapes fixed by the reference harness
//   B = 256 cells, G = 2000 genes, L = 64 latent, C = 128 bins, NCUTS = 500000
// ---------------------------------------------------------------------------

typedef __attribute__((ext_vector_type(2))) float v2f;
typedef __attribute__((ext_vector_type(8))) float v8f;

#define NLAT 128          // number of spline bins (C)
#define LATDIM 64         // latent dim (L)
#define MAXB 256          // cells per batch
#define LOG_NBINS 4.852030263919617f   // log(128)

// ---------------------------------------------------------------------------
// Kernel 1: gather per-gene spline heights: gh[g,c] = spline_heights[genes_oi[g], c]
// ---------------------------------------------------------------------------
__global__ void gather_heights_kernel(const float* __restrict__ spline_heights,
                                      const int* __restrict__ genes_oi,
                                      float* __restrict__ gh, int G) {
    int i = blockIdx.x * blockDim.x + threadIdx.x;
    if (i >= G * NLAT) return;
    int g = i / NLAT, c = i % NLAT;
    gh[i] = spline_heights[(size_t)genes_oi[g] * NLAT + c];
}

// ---------------------------------------------------------------------------
// Kernel 2: histogram cuts by gene (cxg % G) AND Poisson bincount
// ---------------------------------------------------------------------------
__global__ void hist_kernel(const int* __restrict__ cut_cxg,
                            const int* __restrict__ local_cxg,
                            int ncuts, int G,
                            int* __restrict__ gene_counts,
                            int* __restrict__ counts) {
    int i = blockIdx.x * blockDim.x + threadIdx.x;
    if (i >= ncuts) return;
    atomicAdd(&gene_counts[cut_cxg[i] % G], 1);
    atomicAdd(&counts[local_cxg[i]], 1);
}

// ---------------------------------------------------------------------------
// Kernel 3: single-block exclusive scan over gene_counts (G <= 2048)
// ---------------------------------------------------------------------------
__global__ void scan_kernel(const int* __restrict__ gene_counts,
                            int* __restrict__ gene_offsets,
                            int* __restrict__ gene_cursor, int G) {
    __shared__ int s[1024];
    int t = threadIdx.x;                 // 1024 threads
    int i0 = 2 * t, i1 = 2 * t + 1;
    int c0 = (i0 < G) ? gene_counts[i0] : 0;
    int c1 = (i1 < G) ? gene_counts[i1] : 0;
    s[t] = c0 + c1;
    __syncthreads();
    for (int off = 1; off < 1024; off <<= 1) {
        int v = (t >= off) ? s[t - off] : 0;
        __syncthreads();
        s[t] += v;
        __syncthreads();
    }
    int excl = (t > 0) ? s[t - 1] : 0;
    if (i0 < G) { gene_offsets[i0] = excl;      gene_cursor[i0] = excl; }
    if (i1 < G) { gene_offsets[i1] = excl + c0; gene_cursor[i1] = excl + c0; }
    if (t == 1023) gene_offsets[G] = s[1023];
}

// ---------------------------------------------------------------------------
// Kernel 4: scatter cut indices into gene-sorted order
// ---------------------------------------------------------------------------
__global__ void scatter_kernel(const int* __restrict__ cut_cxg,
                               int ncuts, int G,
                               int* __restrict__ gene_cursor,
                               int* __restrict__ sorted_ix) {
    int i = blockIdx.x * blockDim.x + threadIdx.x;
    if (i >= ncuts) return;
    int pos = atomicAdd(&gene_cursor[cut_cxg[i] % G], 1);
    sorted_ix[pos] = i;
}

// ---------------------------------------------------------------------------
// Kernel 5 (main): per-gene fused WMMA GEMM + log-softmax likelihood
//   One workgroup per gene g:
//     delta[b,c] = sum_l latent[b,l] * logit_weight[genes_oi[g], l, c]
//   computed as 16x16 tiles via V_WMMA_F32_16X16X4_F32, accumulated in LDS,
//   then each cut in gene g's bin does a two-pass logsumexp against its
//   (independent) heights row and accumulates the likelihood.
// ---------------------------------------------------------------------------
__global__ void __launch_bounds__(256)
mixture_kernel(const float* __restrict__ latent,
               const float* __restrict__ logit_weight,
               const int* __restrict__ genes_oi,
               const int* __restrict__ gene_offsets,
               const int* __restrict__ sorted_ix,
               const int* __restrict__ cut_cxg,
               const int* __restrict__ cut_gene_ix,
               const float* __restrict__ cut_coord,
               const float* __restrict__ gene_heights,
               float* __restrict__ out,
               int G, int B) {
    __shared__ float lw_s[LATDIM * NLAT];       //  32 KB: B-matrix (64 x 128)
    __shared__ float delta_s[MAXB * NLAT];      // 128 KB: delta tile (256 x 128)
    __shared__ float red[256];

    const int g   = blockIdx.x;
    const int tid = threadIdx.x;

    // ---- stage logit_weight[genes_oi[g]] (64x128 f32) into LDS, float4 ----
    {
        const float4* src = (const float4*)(logit_weight +
                            (size_t)genes_oi[g] * LATDIM * NLAT);
        float4* dst = (float4*)lw_s;
        for (int i = tid; i < (LATDIM * NLAT) / 4; i += 256) dst[i] = src[i];
    }
    __syncthreads();

    // ---- WMMA GEMM: 16 M-tiles x 8 N-tiles x 16 K-steps ----
    const int wave = tid >> 5;
    const int lane = tid & 31;
    const int lm = lane & 15;       // row/col within half-wave
    const int lh = lane >> 4;       // which K/M half

    for (int mi = 0; mi < 2; ++mi) {
        const int M0 = (wave * 2 + mi) * 16;
        // preload A fragments for all 16 K-steps (latent rows, L2-hot 64KB)
        const float* arow = latent + (size_t)(M0 + lm) * LATDIM;
        v2f afrag[16];
        #pragma unroll
        for (int kt = 0; kt < 16; ++kt) {
            const float* p = arow + kt * 4 + 2 * lh;   // 8B aligned
            afrag[kt] = *(const v2f*)p;
        }
        for (int nt = 0; nt < 8; ++nt) {
            const int N0 = nt * 16;
            v8f acc = {};
            #pragma unroll
            for (int kt = 0; kt < 16; ++kt) {
                const int K0 = kt * 4 + 2 * lh;
                v2f bfrag;
                bfrag.x = lw_s[(K0    ) * NLAT + N0 + lm];
                bfrag.y = lw_s[(K0 + 1) * NLAT + N0 + lm];
                acc = __builtin_amdgcn_wmma_f32_16x16x4_f32(
                        false, afrag[kt], false, bfrag,
                        (short)0, acc, false, false);
            }
            // D layout: VGPR r -> row M0 + r + 8*lh, col N0 + lm
            #pragma unroll
            for (int r = 0; r < 8; ++r)
                delta_s[(M0 + r + 8 * lh) * NLAT + N0 + lm] = acc[r];
        }
    }
    __syncthreads();

    // ---- consume cuts belonging to this gene ----
    const int start = gene_offsets[g];
    const int end   = gene_offsets[g + 1];
    float local = 0.0f;
    for (int j = start + tid; j < end; j += 256) {
        const int   i    = sorted_ix[j];
        const int   cxg  = cut_cxg[i];
        const int   b    = cxg / G;                 // cxg % G == g
        const int   g1   = cut_gene_ix[i];          // independent heights gene
        const float co   = cut_coord[i];
        int bin = (int)(co * (float)NLAT);
        bin = bin < 0 ? 0 : (bin > NLAT - 1 ? NLAT - 1 : bin);

        const float* hrow = gene_heights + (size_t)g1 * NLAT;
        const float* drow = delta_s + b * NLAT;

        float mx = -1e30f;
        #pragma unroll 4
        for (int c = 0; c < NLAT; ++c) {
            float v = hrow[c] + drow[c];
            mx = fmaxf(mx, v);
        }
        float se = 0.0f;
        #pragma unroll 4
        for (int c = 0; c < NLAT; ++c)
            se += __expf(hrow[c] + drow[c] - mx);
        const float lse = mx + __logf(se);
        const float vb  = hrow[bin] + drow[bin];
        local += vb - lse + LOG_NBINS;
    }

    // ---- block reduce, negate, accumulate ----
    red[tid] = local;
    __syncthreads();
    for (int s = 128; s > 0; s >>= 1) {
        if (tid < s) red[tid] += red[tid + s];
        __syncthreads();
    }
    if (tid == 0) atomicAdd(out, -red[0]);
}

// ---------------------------------------------------------------------------
// Kernel 6: Poisson fragment-count likelihood over all (b, g)
// ---------------------------------------------------------------------------
__global__ void __launch_bounds__(256)
poisson_kernel(const float* __restrict__ latent,
               const float* __restrict__ rho_weight,
               const float* __restrict__ rho_bias,
               const int* __restrict__ genes_oi,
               const int* __restrict__ cells_oi,
               const int* __restrict__ libsize,
               const int* __restrict__ counts,
               float* __restrict__ out,
               int B, int G) {
    __shared__ float red[256];
    const int tid = threadIdx.x;
    const int idx = blockIdx.x * 256 + tid;
    float local = 0.0f;
    if (idx < B * G) {
        const int b = idx / G;
        const int g = idx % G;
        const int gg = genes_oi[g];
        const float* lrow = latent + (size_t)b * LATDIM;
        const float* wrow = rho_weight + (size_t)gg * LATDIM;
        float dot = 0.0f;
        #pragma unroll 8
        for (int l = 0; l < LATDIM; ++l) dot = fmaf(lrow[l], wrow[l], dot);
        const float rate = rho_bias[gg] * __expf(dot) *
                           (float)libsize[cells_oi[b]];
        const float cnt = (float)counts[idx];
        local = cnt * __logf(rate) - rate - lgammaf(cnt + 1.0f);
    }
    red[tid] = local;
    __syncthreads();
    for (int s = 128; s > 0; s >>= 1) {
        if (tid < s) red[tid] += red[tid + s];
        __syncthreads();
    }
    if (tid == 0) atomicAdd(out, -red[0]);
}

// ---------------------------------------------------------------------------
// Host launcher (graph-capture safe: only async ops on `stream`)
// ---------------------------------------------------------------------------
extern "C" void kernel_launch(void* const* d_in, const int* in_sizes, int n_in,
                              void* d_out, int out_size, void* d_ws, size_t ws_size,
                              hipStream_t stream) {
    const float* latent        = (const float*)d_in[0];
    const int*   genes_oi      = (const int*)  d_in[1];
    const int*   cells_oi      = (const int*)  d_in[2];
    const float* cut_coord     = (const float*)d_in[3];
    const int*   cut_cxg       = (const int*)  d_in[4];
    const int*   cut_gene_ix   = (const int*)  d_in[5];
    const int*   local_cxg     = (const int*)  d_in[6];
    const float* logit_weight  = (const float*)d_in[7];
    const float* rho_weight    = (const float*)d_in[8];
    const float* rho_bias      = (const float*)d_in[9];
    const int*   libsize       = (const int*)  d_in[10];
    const float* spline_hts    = (const float*)d_in[11];

    const int B  = in_sizes[0] / LATDIM;   // 256
    const int G  = in_sizes[1];            // 2000
    const int NC = in_sizes[3];            // 500000

    // ---- workspace layout (all 256B aligned) ----
    char* ws = (char*)d_ws;
    size_t off = 0;
    int* counts       = (int*)(ws + off); off += (size_t)B * G * sizeof(int);
    off = (off + 255) & ~(size_t)255;
    size_t zero_end   = off + 8192;        // gene_counts zeroed together with counts
    int* gene_counts  = (int*)(ws + off); off += 8192;
    int* gene_offsets = (int*)(ws + off); off += 8192;
    int* gene_cursor  = (int*)(ws + off); off += 8192;
    int* sorted_ix    = (int*)(ws + off); off += ((size_t)NC * 4 + 255) & ~(size_t)255;
    float* gene_hts   = (float*)(ws + off); off += (size_t)G * NLAT * sizeof(float);
    (void)ws_size; (void)n_in; (void)out_size;

    float* out = (float*)d_out;
    hipMemsetAsync(out, 0, sizeof(float), stream);
    hipMemsetAsync(d_ws, 0, zero_end, stream);

    gather_heights_kernel<<<(G * NLAT + 255) / 256, 256, 0, stream>>>(
        spline_hts, genes_oi, gene_hts, G);

    hist_kernel<<<(NC + 255) / 256, 256, 0, stream>>>(
        cut_cxg, local_cxg, NC, G, gene_counts, counts);

    scan_kernel<<<1, 1024, 0, stream>>>(gene_counts, gene_offsets, gene_cursor, G);

    scatter_kernel<<<(NC + 255) / 256, 256, 0, stream>>>(
        cut_cxg, NC, G, gene_cursor, sorted_ix);

    mixture_kernel<<<G, 256, 0, stream>>>(
        latent, logit_weight, genes_oi, gene_offsets, sorted_ix,
        cut_cxg, cut_gene_ix, cut_coord, gene_hts, out, G, B);

    poisson_kernel<<<(B * G + 255) / 256, 256, 0, stream>>>(
        latent, rho_weight, rho_bias, genes_oi, cells_oi, libsize,
        counts, out, B, G);
}